// ChamferLossSplitPID_68393059221687
// MI455X (gfx1250) — compile-verified
//
#include <hip/hip_runtime.h>
#include <math.h>

// ChamferLossSplitPID for MI455X (gfx1250, wave32).
// One workgroup (256 threads = 8 waves) per batch element (B=512).
// Cross term computed as 16x16 tiles with V_WMMA_F32_16X16X4_F32 (K = D = 4).
// target is staged into LDS pre-scaled by -2, and C = x2[row] + y2[col] is fed
// through the WMMA accumulator, so the WMMA output is the squared distance
// directly: D = x2 + y2 - 2*xy. Phase 2 is fully unrolled straight-line code
// (32 WMMAs per wave), branchless masked mins, ds_min_u32 on float bit patterns.
// (This is the Round-3 formulation, which produced the best instruction mix:
// total=3891, wmma=32, no divergent exec-mask branches in the element loop.)

typedef __attribute__((ext_vector_type(2))) float v2f;
typedef __attribute__((ext_vector_type(8))) float v8f;

#define BIGV 1.0e9f

__device__ __forceinline__ float waveSum32(float v) {
#pragma unroll
  for (int off = 16; off > 0; off >>= 1) v += __shfl_down(v, off, 32);
  return v;
}

__global__ __launch_bounds__(256) void chamfer_pid_kernel(
    const float* __restrict__ target, const float* __restrict__ reco,
    const int* __restrict__ in_pid, const int* __restrict__ out_pid,
    float* __restrict__ zero_out,  // d_out + 1, [B]
    float* __restrict__ nz_ws,     // workspace, [B]
    int B) {
  __shared__ float sT[256 * 4];  // holds -2 * target
  __shared__ float sR[256 * 4];
  __shared__ float sX2[256];
  __shared__ float sY2[256];
  __shared__ float sNX[256];
  __shared__ float sNY[256];
  __shared__ int sPX[256];
  __shared__ int sPY[256];
  __shared__ unsigned sMinXY[256];
  __shared__ unsigned sMinYX[256];
  __shared__ float sPart[20][8];

  const int b = blockIdx.x;
  const int t = threadIdx.x;
  const int lane = t & 31;
  const int wave = __builtin_amdgcn_readfirstlane(t >> 5);  // scalar

  // ---- Phase 1: stage batch into LDS, precompute norms ----
  const float4 tv = reinterpret_cast<const float4*>(target)[b * 256 + t];
  const float4 rv = reinterpret_cast<const float4*>(reco)[b * 256 + t];
  float4 tscaled;
  tscaled.x = -2.0f * tv.x;
  tscaled.y = -2.0f * tv.y;
  tscaled.z = -2.0f * tv.z;
  tscaled.w = -2.0f * tv.w;
  reinterpret_cast<float4*>(sT)[t] = tscaled;
  reinterpret_cast<float4*>(sR)[t] = rv;
  const float x2 = tv.x * tv.x + tv.y * tv.y + tv.z * tv.z + tv.w * tv.w;
  const float y2 = rv.x * rv.x + rv.y * rv.y + rv.z * rv.z + rv.w * rv.w;
  sX2[t] = x2;
  sY2[t] = y2;
  sNX[t] = sqrtf(x2);
  sNY[t] = sqrtf(y2);
  sPX[t] = in_pid[b * 256 + t];
  sPY[t] = out_pid[b * 256 + t];
  sMinXY[t] = __float_as_uint(BIGV);
  sMinYX[t] = __float_as_uint(BIGV);
  __syncthreads();

  // ---- Phase 2: straight-line WMMA tiles, D = x2 + y2 - 2*xy via accumulator ----
  // A fragment (16x4, MxK): lanes 0-15 -> K=0,1 of row base+lane; lanes 16-31 -> K=2,3.
  // B fragment (4x16, KxN): identical addressing over reco rows (=> reco^T).
  // C/D layout: VGPR r, lanes 0-15 -> (M=r, N=lane); lanes 16-31 -> (M=r+8, N=lane-16).
  const int l15 = lane & 15;
  const int koff = (lane >> 4) << 1;   // 0 or 2
  const int mbase = (lane >> 4) << 3;  // 0 or 8

#pragma unroll
  for (int tiSub = 0; tiSub < 2; ++tiSub) {
    const int ti = wave * 2 + tiSub;
    const int rowA = ti * 16 + l15;
    const v2f a = *reinterpret_cast<const v2f*>(&sT[rowA * 4 + koff]);

    // Row-invariant data for this tile-row (8 rows per lane half).
    float x2row[8];
    int pidrow[8];
    float rowMin[8];
#pragma unroll
    for (int r = 0; r < 8; ++r) {
      const int rowN = ti * 16 + mbase + r;
      x2row[r] = sX2[rowN];
      pidrow[r] = sPX[rowN];
      rowMin[r] = BIGV;
    }

#pragma unroll
    for (int tj = 0; tj < 16; ++tj) {
      const int rowB = tj * 16 + l15;
      const v2f bb = *reinterpret_cast<const v2f*>(&sR[rowB * 4 + koff]);
      const int col = tj * 16 + l15;
      const int py = sPY[col];
      const float cy2 = sY2[col];
      const bool pyok = (py != 0);

      // Accumulator carries x2[row] + y2[col]; WMMA adds -2*xy.
      v8f cin;
#pragma unroll
      for (int r = 0; r < 8; ++r) cin[r] = x2row[r] + cy2;
      const v8f d2v = __builtin_amdgcn_wmma_f32_16x16x4_f32(
          false, a, false, bb, (short)0, cin, false, false);

      float colMin = BIGV;
#pragma unroll
      for (int r = 0; r < 8; ++r) {
        const float d = __builtin_amdgcn_sqrtf(fmaxf(d2v[r], 0.0f));
        const bool m = pyok && (pidrow[r] == py);
        const float dm = m ? d : BIGV;  // branchless
        rowMin[r] = fminf(rowMin[r], dm);
        colMin = fminf(colMin, dm);
      }
      atomicMin(&sMinYX[col], __float_as_uint(colMin));  // ds_min_u32
    }

#pragma unroll
    for (int r = 0; r < 8; ++r)
      atomicMin(&sMinXY[ti * 16 + mbase + r], __float_as_uint(rowMin[r]));
  }
  __syncthreads();

  // ---- Phase 3: per-PID sums, deterministic tree reduction ----
  const float minxy = __uint_as_float(sMinXY[t]);
  const float minyx = __uint_as_float(sMinYX[t]);
  const float nx = sNX[t], ny = sNY[t];
  const int px = sPX[t], py = sPY[t];

  float comp[20];
#pragma unroll
  for (int p = 1; p <= 3; ++p) {
    const int base = (p - 1) * 6;
    const bool ix = (px == p);
    const bool iy = (py == p);
    comp[base + 0] = ix ? 1.0f : 0.0f;   // cx
    comp[base + 1] = iy ? 1.0f : 0.0f;   // cy
    comp[base + 2] = ix ? minxy : 0.0f;  // sum of per-x mins
    comp[base + 3] = iy ? minyx : 0.0f;  // sum of per-y mins
    comp[base + 4] = ix ? nx : 0.0f;     // sum norm_x
    comp[base + 5] = iy ? ny : 0.0f;     // sum norm_y
  }
  comp[18] = (py == 0) ? 1.0f : 0.0f;
  comp[19] = (py == 0) ? ny : 0.0f;

#pragma unroll
  for (int i = 0; i < 20; ++i) {
    const float s = waveSum32(comp[i]);
    if (lane == 0) sPart[i][wave] = s;
  }
  __syncthreads();

  if (t == 0) {
    float tot[20];
#pragma unroll
    for (int i = 0; i < 20; ++i) {
      float s = 0.0f;
#pragma unroll
      for (int w = 0; w < 8; ++w) s += sPart[i][w];
      tot[i] = s;
    }
    float nz = 0.0f;
#pragma unroll
    for (int p = 0; p < 3; ++p) {
      const int base = p * 6;
      const float cx = tot[base + 0], cyc = tot[base + 1];
      const float n_in = fmaxf(cx, 1.0f);
      const float n_out = fmaxf(cyc, 1.0f);
      const float both = 0.5f * (tot[base + 2] / n_out + tot[base + 3] / n_in);
      const float only_x = tot[base + 4] / n_in;
      const float only_y = tot[base + 5] / n_out;
      nz += (cyc == 0.0f) ? only_x : ((cx == 0.0f) ? only_y : both);
    }
    nz_ws[b] = nz;
    zero_out[b] = (tot[19] / fmaxf(tot[18], 1.0f)) / (float)B;
  }
}

__global__ __launch_bounds__(256) void chamfer_final_reduce(
    const float* __restrict__ nz_ws, float* __restrict__ out_scalar, int B) {
  __shared__ float s[256];
  const int t = threadIdx.x;
  float v = 0.0f;
  for (int i = t; i < B; i += 256) v += nz_ws[i];
  s[t] = v;
  __syncthreads();
  for (int off = 128; off > 0; off >>= 1) {
    if (t < off) s[t] += s[t + off];
    __syncthreads();
  }
  if (t == 0) out_scalar[0] = s[0] / (float)B;
}

extern "C" void kernel_launch(void* const* d_in, const int* in_sizes, int n_in,
                              void* d_out, int out_size, void* d_ws,
                              size_t ws_size, hipStream_t stream) {
  const float* target = (const float*)d_in[0];
  const float* reco = (const float*)d_in[1];
  const int* in_pid = (const int*)d_in[2];
  const int* out_pid = (const int*)d_in[3];
  float* out = (float*)d_out;   // [0] = scalar loss, [1..B] = zero_per_batch
  float* nz_ws = (float*)d_ws;  // [B] floats of scratch

  const int N = 256;
  const int B = in_sizes[0] / (N * 4);

  chamfer_pid_kernel<<<B, 256, 0, stream>>>(target, reco, in_pid, out_pid,
                                            out + 1, nz_ws, B);
  chamfer_final_reduce<<<1, 256, 0, stream>>>(nz_ws, out, B);
}